// TimeMultiHeadAttention_74844100100620
// MI455X (gfx1250) — compile-verified
//
#include <hip/hip_runtime.h>
#include <math.h>

#define D_MODEL   256
#define NUM_HEADS 8
#define DK        32
#define B_SZ      4
#define T_SZ      2048

typedef float v2f __attribute__((ext_vector_type(2)));
typedef float v8f __attribute__((ext_vector_type(8)));

// D = A(16x4 f32) * B(4x16 f32) + C(16x16 f32)  -- V_WMMA_F32_16X16X4_F32
__device__ __forceinline__ v8f wmma4(v2f a, v2f b, v8f c) {
    return __builtin_amdgcn_wmma_f32_16x16x4_f32(
        /*neg_a=*/false, a, /*neg_b=*/false, b,
        /*c_mod=*/(short)0, c, /*reuse_a=*/false, /*reuse_b=*/false);
}

// ---------------------------------------------------------------------------
// Kernel 1: fused Q/K/V projection.  out = x @ W^T + b.
// Q,K stored [B,H,T,DK]; V stored transposed [B,H,DK,T] so the attention
// PV stage can load its B-fragment k-pairs with contiguous b64 loads.
// One wave per 16(token) x 16(channel) tile of one of the 3 projections.
// ---------------------------------------------------------------------------
__global__ __launch_bounds__(256)
void qkv_proj_kernel(const float* __restrict__ x,
                     const float* __restrict__ Wq, const float* __restrict__ bq,
                     const float* __restrict__ Wk, const float* __restrict__ bk,
                     const float* __restrict__ Wv, const float* __restrict__ bv,
                     float* __restrict__ QKV)
{
    const int lane = threadIdx.x & 31;
    const int w    = threadIdx.x >> 5;
    const int tile = blockIdx.x * 8 + w;

    const int NT  = D_MODEL / 16;        // 16 n-tiles
    const int MT  = (B_SZ * T_SZ) / 16;  // 512 m-tiles
    const int mat = tile / (MT * NT);    // 0:Q 1:K 2:V
    const int rem = tile % (MT * NT);
    const int mt  = rem / NT;
    const int nt  = rem % NT;

    const float* W  = (mat == 0) ? Wq : (mat == 1) ? Wk : Wv;
    const float* bs = (mat == 0) ? bq : (mat == 1) ? bk : bv;

    const int r15 = lane & 15;           // A row / B,C,D column within tile
    const int grp = lane >> 4;           // k-pair group select
    const int n   = nt * 16 + r15;       // output channel this lane owns
    const int m0  = mt * 16;

    const float* xr = x + (size_t)(m0 + r15) * D_MODEL;  // A row
    const float* wr = W + (size_t)n * D_MODEL;           // B column = W row

    v8f acc = {};
    for (int k0 = 0; k0 < D_MODEL; k0 += 4) {
        const int ka = k0 + grp * 2;
        v2f a = *(const v2f*)(xr + ka);
        v2f b = *(const v2f*)(wr + ka);
        acc = wmma4(a, b, acc);
    }

    const float bias = bs[n];
    const int h = n >> 5, d = n & 31;
    float* dst = QKV + (size_t)mat * ((size_t)B_SZ * NUM_HEADS * T_SZ * DK);
#pragma unroll
    for (int m = 0; m < 8; ++m) {
        const int tl = m0 + m + 8 * grp;           // linear token (b*T + t)
        const int b_ = tl / T_SZ, t_ = tl % T_SZ;
        const float v = acc[m] + bias;
        if (mat == 2) {  // V transposed: [B,H,DK,T]
            dst[(((size_t)(b_ * NUM_HEADS + h)) * DK + d) * T_SZ + t_] = v;
        } else {         // Q,K: [B,H,T,DK]
            dst[(((size_t)(b_ * NUM_HEADS + h)) * T_SZ + t_) * DK + d] = v;
        }
    }
}

// ---------------------------------------------------------------------------
// Attention core: causal attention with temporal/oc prior, flash-style
// online softmax.  Templated on BETA_ONE so the |beta|==1 common case skips
// the per-element __powf (log+mul+exp) entirely; the branch selecting the
// instantiation is wave-uniform (scalar).
// ---------------------------------------------------------------------------
template <bool BETA_ONE>
__device__ __forceinline__ void attn_core(
    const float* __restrict__ Q, const float* __restrict__ K,
    const float* __restrict__ Vt, const float* __restrict__ oc,
    float* __restrict__ ctx, float* __restrict__ pl,
    int b, int h, int qt, int lane,
    float eta, float absbeta, float alpha)
{
    const int col = lane & 15;
    const int grp = lane >> 4;
    const int q0  = qt * 16;

    const float scale   = 0.17677669529663687f;  // 1/sqrt(32)
    const float NEG_BIG = -3.4e38f;

    const size_t bh  = (size_t)(b * NUM_HEADS + h) * T_SZ;
    const size_t bhd = (size_t)(b * NUM_HEADS + h) * DK;

    // Preload this lane's Q-row A-fragments (row = col, k split by lane group)
    const float* qp = Q + (bh + q0 + col) * DK;
    v2f qa[8];
#pragma unroll
    for (int kc = 0; kc < 8; ++kc) qa[kc] = *(const v2f*)(qp + kc * 4 + grp * 2);

    // oc for the 8 q-rows this lane's accumulator covers
    float qoc0[8], qoc1[8], qoc2[8];
#pragma unroll
    for (int m = 0; m < 8; ++m) {
        const int qq = q0 + 8 * grp + m;
        const float* op = oc + ((size_t)b * T_SZ + qq) * 3;
        qoc0[m] = op[0]; qoc1[m] = op[1]; qoc2[m] = op[2];
    }

    // Vt rows this lane streams (d = col and d = col+16), contiguous in k
    const float* vt0 = Vt + (bhd + col) * T_SZ;
    const float* vt1 = vt0 + 16 * T_SZ;

    float Mrow[8], Lrow[8];
#pragma unroll
    for (int m = 0; m < 8; ++m) { Mrow[m] = NEG_BIG; Lrow[m] = 0.f; }
    v8f O0 = {}, O1 = {};

    for (int kt = 0; kt <= qt; ++kt) {
        const int k0   = kt * 16;
        const int kcol = k0 + col;       // the key index this lane owns
        const float* kp = K + (bh + kcol) * DK;

        // pull next tile's lines toward the WGP while we do softmax VALU work
        if (kt < qt) {
            __builtin_prefetch(kp + 16 * DK, 0, 0);     // next K rows
            __builtin_prefetch(vt0 + k0 + 16, 0, 0);    // next Vt chunk (d-tile 0)
            __builtin_prefetch(vt1 + k0 + 16, 0, 0);    // next Vt chunk (d-tile 1)
        }

        // scores tile: S = Q * K^T (B fragment: column = K row kcol)
        v8f s = {};
#pragma unroll
        for (int kc = 0; kc < 8; ++kc) {
            v2f bb = *(const v2f*)(kp + kc * 4 + grp * 2);
            s = wmma4(qa[kc], bb, s);
        }

        const float* kop = oc + ((size_t)b * T_SZ + kcol) * 3;
        const float koc0 = kop[0], koc1 = kop[1], koc2 = kop[2];

#pragma unroll
        for (int m = 0; m < 8; ++m) {
            const int qrow = q0 + 8 * grp + m;
            if (kcol > qrow) {
                s[m] = NEG_BIG;          // causal mask
            } else {
                const float dist = (float)(qrow - kcol);
                const float powd = BETA_ONE ? dist : __powf(dist, absbeta);
                const float td   = eta / (1.f + powd);
                const float d0 = qoc0[m] - koc0;
                const float d1 = qoc1[m] - koc1;
                const float d2 = qoc2[m] - koc2;
                const float pr = alpha * td * __expf(-(d0*d0 + d1*d1 + d2*d2));
                s[m] = s[m] * scale * pr;
            }
        }

        // online softmax: row reductions across the 16 lanes of each half
#pragma unroll
        for (int m = 0; m < 8; ++m) {
            float v = s[m];
            v = fmaxf(v, __shfl_xor(v, 1));
            v = fmaxf(v, __shfl_xor(v, 2));
            v = fmaxf(v, __shfl_xor(v, 4));
            v = fmaxf(v, __shfl_xor(v, 8));
            const float mnew = fmaxf(Mrow[m], v);
            const float sc   = __expf(Mrow[m] - mnew);
            const float p    = __expf(s[m] - mnew);
            float su = p;
            su += __shfl_xor(su, 1);
            su += __shfl_xor(su, 2);
            su += __shfl_xor(su, 4);
            su += __shfl_xor(su, 8);
            Lrow[m] = Lrow[m] * sc + su;
            Mrow[m] = mnew;
            O0[m] *= sc;
            O1[m] *= sc;
            pl[(8 * grp + m) * 16 + col] = p;   // P row-major into LDS
        }

        // O += P @ V   (A = P from LDS; B = Vt rows, contiguous k-pairs)
#pragma unroll
        for (int kc = 0; kc < 4; ++kc) {
            const int kk = kc * 4 + grp * 2;
            v2f pa;
            pa.x = pl[col * 16 + kk];
            pa.y = pl[col * 16 + kk + 1];
            v2f b0 = *(const v2f*)(vt0 + k0 + kk);
            v2f b1 = *(const v2f*)(vt1 + k0 + kk);
            O0 = wmma4(pa, b0, O0);
            O1 = wmma4(pa, b1, O1);
        }
    }

    // normalize and write ctx in [B,T,d_model] layout for the out-projection
#pragma unroll
    for (int m = 0; m < 8; ++m) {
        const float inv = 1.f / Lrow[m];
        const int qq = q0 + 8 * grp + m;
        float* cp = ctx + ((size_t)b * T_SZ + qq) * D_MODEL + h * DK;
        cp[col]      = O0[m] * inv;
        cp[16 + col] = O1[m] * inv;
    }
}

__global__ __launch_bounds__(256)
void attn_kernel(const float* __restrict__ QKV,
                 const float* __restrict__ oc,
                 const float* __restrict__ etap,
                 const float* __restrict__ betap,
                 const float* __restrict__ alphap,
                 float* __restrict__ ctx)
{
    __shared__ float plds[8][256];       // per-wave 16x16 P staging tile

    const int lane = threadIdx.x & 31;
    const int w    = threadIdx.x >> 5;
    const int wid  = blockIdx.x * 8 + w;

    const int QT = T_SZ / 16;
    const int qt = wid % QT;
    const int h  = (wid / QT) % NUM_HEADS;
    const int b  = wid / (QT * NUM_HEADS);

    const float eta     = etap[0];
    const float absbeta = fabsf(betap[0]);
    const float alpha   = alphap[0];

    const size_t HT = (size_t)B_SZ * NUM_HEADS * T_SZ * DK;
    const float* Q  = QKV;
    const float* K  = QKV + HT;
    const float* Vt = QKV + 2 * HT;      // [B,H,DK,T]

    float* pl = &plds[w][0];

    if (absbeta == 1.0f) {               // wave-uniform scalar branch
        attn_core<true >(Q, K, Vt, oc, ctx, pl, b, h, qt, lane, eta, absbeta, alpha);
    } else {
        attn_core<false>(Q, K, Vt, oc, ctx, pl, b, h, qt, lane, eta, absbeta, alpha);
    }
}

// ---------------------------------------------------------------------------
// Kernel 3: output projection  out = ctx @ Wo^T + bo
// ---------------------------------------------------------------------------
__global__ __launch_bounds__(256)
void out_proj_kernel(const float* __restrict__ ctx,
                     const float* __restrict__ Wo, const float* __restrict__ bo,
                     float* __restrict__ out)
{
    const int lane = threadIdx.x & 31;
    const int w    = threadIdx.x >> 5;
    const int tile = blockIdx.x * 8 + w;

    const int NT = D_MODEL / 16;
    const int mt = tile / NT;
    const int nt = tile % NT;

    const int r15 = lane & 15;
    const int grp = lane >> 4;
    const int n   = nt * 16 + r15;
    const int m0  = mt * 16;

    const float* xr = ctx + (size_t)(m0 + r15) * D_MODEL;
    const float* wr = Wo + (size_t)n * D_MODEL;

    v8f acc = {};
    for (int k0 = 0; k0 < D_MODEL; k0 += 4) {
        const int ka = k0 + grp * 2;
        v2f a = *(const v2f*)(xr + ka);
        v2f b = *(const v2f*)(wr + ka);
        acc = wmma4(a, b, acc);
    }

    const float bias = bo[n];
#pragma unroll
    for (int m = 0; m < 8; ++m) {
        const int tl = m0 + m + 8 * grp;
        out[(size_t)tl * D_MODEL + n] = acc[m] + bias;
    }
}

// ---------------------------------------------------------------------------
extern "C" void kernel_launch(void* const* d_in, const int* in_sizes, int n_in,
                              void* d_out, int out_size, void* d_ws, size_t ws_size,
                              hipStream_t stream) {
    (void)in_sizes; (void)n_in; (void)out_size; (void)ws_size;

    const float* x     = (const float*)d_in[0];
    const float* oc    = (const float*)d_in[1];
    const float* Wq    = (const float*)d_in[2];
    const float* bq    = (const float*)d_in[3];
    const float* Wk    = (const float*)d_in[4];
    const float* bk    = (const float*)d_in[5];
    const float* Wv    = (const float*)d_in[6];
    const float* bv    = (const float*)d_in[7];
    const float* Wo    = (const float*)d_in[8];
    const float* bo    = (const float*)d_in[9];
    const float* eta   = (const float*)d_in[10];
    const float* beta  = (const float*)d_in[11];
    const float* alpha = (const float*)d_in[12];
    float* out = (float*)d_out;

    float* ws  = (float*)d_ws;
    float* QKV = ws;                                               // 3*B*H*T*DK floats
    float* ctx = ws + 3 * (size_t)B_SZ * NUM_HEADS * T_SZ * DK;    // B*T*D_MODEL floats

    {   // 3 * 512 * 16 = 24576 tiles, 8 waves/block
        const int tiles = 3 * (B_SZ * T_SZ / 16) * (D_MODEL / 16);
        qkv_proj_kernel<<<tiles / 8, 256, 0, stream>>>(x, Wq, bq, Wk, bk, Wv, bv, QKV);
    }
    {   // 4*8*128 = 4096 waves, 8 waves/block
        const int waves = B_SZ * NUM_HEADS * (T_SZ / 16);
        attn_kernel<<<waves / 8, 256, 0, stream>>>(QKV, oc, eta, beta, alpha, ctx);
    }
    {   // 512 * 16 = 8192 tiles
        const int tiles = (B_SZ * T_SZ / 16) * (D_MODEL / 16);
        out_proj_kernel<<<tiles / 8, 256, 0, stream>>>(ctx, Wo, bo, out);
    }
}